// QSDABlock_17549236371754
// MI455X (gfx1250) — compile-verified
//
#include <hip/hip_runtime.h>
#include <math.h>

// Problem dims (fixed by reference)
#define BB   4
#define NN   4096
#define RR   (BB * NN)      // 16384 rows
#define DD   1024
#define DH   256
#define SS   256
#define HIDN 32
#define FF   4096

typedef float v2f __attribute__((ext_vector_type(2)));
typedef float v8f __attribute__((ext_vector_type(8)));
typedef float f4  __attribute__((ext_vector_type(4)));

// ---------------- WMMA helpers (fp32 matrix path, 16x16x4) ----------------
__device__ __forceinline__ v8f wmma4(v2f a, v2f b, v8f c) {
  return __builtin_amdgcn_wmma_f32_16x16x4_f32(false, a, false, b, (short)0, c,
                                               false, false);
}

// A fragment (16x4 MxK): lane L: m=L&15; VGPR0=A[m][k0+2*(L>=16)], VGPR1=+1
__device__ __forceinline__ v2f fragA(const float* base, int lda, int k0) {
  int lane = threadIdx.x & 31;
  int m = lane & 15;
  int kh = k0 + ((lane >> 4) << 1);
  const float* p = base + m * lda + kh;
  v2f a;
  a.x = p[0];
  a.y = p[1];
  return a;
}

// B fragment (4x16 KxN) row-major B[k][n]; base pre-offset to (k0,n0)
__device__ __forceinline__ v2f fragB(const float* base, int ldb) {
  int lane = threadIdx.x & 31;
  int n = lane & 15;
  int kh = (lane >> 4) << 1;
  v2f b;
  b.x = base[kh * ldb + n];
  b.y = base[(kh + 1) * ldb + n];
  return b;
}

// Async global->LDS 16B copy (GLOBAL_LOAD_ASYNC_TO_LDS_B128, ASYNCcnt).
// Flat LDS address low 32 bits == LDS offset (aperture in high 32 bits).
__device__ __forceinline__ void async_ld_b128(void* lds_dst, const void* gsrc) {
  unsigned off = (unsigned)(size_t)lds_dst;
  asm volatile("global_load_async_to_lds_b128 %0, %1, off" ::"v"(off),
               "v"(gsrc)
               : "memory");
}
__device__ __forceinline__ void wait_async0() {
  asm volatile("s_wait_asynccnt 0" ::: "memory");
}

__device__ __forceinline__ float wred32(float x) {
#pragma unroll
  for (int o = 16; o > 0; o >>= 1) x += __shfl_xor(x, o, 32);
  return x;
}
__device__ __forceinline__ float hred16(float x) {  // reduce within half-wave
#pragma unroll
  for (int o = 8; o > 0; o >>= 1) x += __shfl_xor(x, o, 32);
  return x;
}
__device__ __forceinline__ float sigmoidf_(float x) {
  return 1.f / (1.f + expf(-x));
}

// ---------------- K0: ctx = mean_n h ; gamma = gate(ctx) ----------------
__global__ __launch_bounds__(1024) void ctx_gamma_kernel(
    const float* __restrict__ h, const float* __restrict__ gw1,
    const float* __restrict__ gb1, const float* __restrict__ gg,
    const float* __restrict__ gbeta, const float* __restrict__ gw2,
    const float* __restrict__ gb2, float* __restrict__ gamma) {
  __shared__ float sctx[DD];
  int b = blockIdx.x;
  int d = threadIdx.x;
  const float* hb = h + (size_t)b * NN * DD;
  float s = 0.f;
  for (int n = 0; n < NN; ++n) s += hb[n * DD + d];
  sctx[d] = s * (1.0f / NN);
  __syncthreads();
  if (threadIdx.x < 32) {  // wave 0 does the 32-wide gate
    int j = threadIdx.x;
    float t = gb1[j];
    for (int k = 0; k < DD; ++k) t = fmaf(sctx[k], gw1[k * HIDN + j], t);
    float mu = wred32(t) * (1.f / HIDN);
    float dv = t - mu;
    float var = wred32(dv * dv) * (1.f / HIDN);
    float tn = dv * rsqrtf(var + 1e-5f) * gg[j] + gbeta[j];
    float si = tn * sigmoidf_(tn);
    float y = wred32(si * gw2[j]) + gb2[0];
    if (j == 0) gamma[b] = sigmoidf_(y);
  }
}

// ---------------- K1: p[row] = 0.95*gate(h_row)*(1-gamma[b]) ----------------
__global__ __launch_bounds__(256) void gate_p_kernel(
    const float* __restrict__ h, const float* __restrict__ uw1,
    const float* __restrict__ ub1, const float* __restrict__ ug,
    const float* __restrict__ ubeta, const float* __restrict__ uw2,
    const float* __restrict__ ub2, const float* __restrict__ gamma,
    float* __restrict__ P) {
  int row = (blockIdx.x * 256 + threadIdx.x) >> 5;  // one wave per row
  int j = threadIdx.x & 31;                         // lane = hidden unit
  const float* hr = h + (size_t)row * DD;
  float t = ub1[j];
  for (int k = 0; k < DD; ++k) t = fmaf(hr[k], uw1[k * HIDN + j], t);
  float mu = wred32(t) * (1.f / HIDN);
  float dv = t - mu;
  float var = wred32(dv * dv) * (1.f / HIDN);
  float tn = dv * rsqrtf(var + 1e-5f) * ug[j] + ubeta[j];
  float si = tn * sigmoidf_(tn);
  float y = wred32(si * uw2[j]) + ub2[0];
  float p0 = 0.95f * sigmoidf_(y);
  float g = gamma[row >> 12];  // N = 4096 rows per batch
  if (j == 0) P[row] = p0 * (1.f - g);
}

// ------- K2: normalize memory rows, store transposed (dH x S) for GEMM -------
__global__ __launch_bounds__(256) void mnorm_kernel(
    const float* __restrict__ m_real, const float* __restrict__ m_imag,
    float* __restrict__ MrT, float* __restrict__ MiT) {
  int s = (blockIdx.x * 256 + threadIdx.x) >> 5;  // one wave per memory slot
  int lane = threadIdx.x & 31;
  const float* mr = m_real + s * DH;
  const float* mi = m_imag + s * DH;
  float ss = 0.f;
  for (int d = lane; d < DH; d += 32) {
    float a = mr[d], c = mi[d];
    ss = fmaf(a, a, fmaf(c, c, ss));
  }
  ss = wred32(ss);
  float inv = 1.f / fmaxf(sqrtf(ss), 1e-12f);
  for (int d = lane; d < DH; d += 32) {
    MrT[d * SS + s] = mr[d] * inv;
    MiT[d * SS + s] = mi[d] * inv;
  }
}

// ---------------- K3: PR = h@Wr+br, PI = h@Wi+bi (WMMA) ----------------
#define ENC_LDA 260  // 260 % 64 == 4 -> conflict-free A-fragment reads
__global__ __launch_bounds__(256) void enc_kernel(
    const float* __restrict__ h, const float* __restrict__ Wr,
    const float* __restrict__ br, const float* __restrict__ Wi,
    const float* __restrict__ bi, float* __restrict__ PR,
    float* __restrict__ PI) {
  __shared__ float sA[16 * ENC_LDA];
  int row0 = blockIdx.x * 16;
  int w = threadIdx.x >> 5;
  int lane = threadIdx.x & 31;
  int mat = w & 1;                // even waves -> PR, odd -> PI
  int colbase = (w >> 1) * 64;    // 4 col-tiles per wave
  const float* W = mat ? Wi : Wr;
  const float* bias = mat ? bi : br;
  v8f acc[4] = {};
  for (int kc = 0; kc < DD; kc += 256) {
    __syncthreads();
    // stage 16x256 chunk of h directly into LDS (async, no VGPR round-trip)
    for (int i = 0; i < 4; ++i) {
      int e = (threadIdx.x << 2) + i * 1024;
      int r = e >> 8, c = e & 255;
      async_ld_b128(&sA[r * ENC_LDA + c], &h[(row0 + r) * DD + kc + c]);
    }
    if (kc + 256 < DD)  // hint next chunk into cache (global_prefetch_b8)
      __builtin_prefetch(&h[(row0 + (threadIdx.x & 15)) * DD + kc + 256], 0, 1);
    wait_async0();
    __syncthreads();
    for (int kk = 0; kk < 256; kk += 4) {
      v2f a = fragA(sA, ENC_LDA, kk);
#pragma unroll
      for (int t = 0; t < 4; ++t) {
        v2f b = fragB(W + (kc + kk) * DH + colbase + t * 16, DH);
        acc[t] = wmma4(a, b, acc[t]);
      }
    }
  }
  int n = lane & 15;
  int mb = (lane >> 4) << 3;
  float* OUT = mat ? PI : PR;
#pragma unroll
  for (int t = 0; t < 4; ++t) {
    int col = colbase + t * 16 + n;
    float bv = bias[col];
#pragma unroll
    for (int r = 0; r < 8; ++r)
      OUT[(row0 + mb + r) * DH + col] = acc[t][r] + bv;
  }
}

// ---------------- K4: psi row normalization ----------------
__global__ __launch_bounds__(256) void psinorm_kernel(float* __restrict__ PR,
                                                      float* __restrict__ PI) {
  int row = (blockIdx.x * 256 + threadIdx.x) >> 5;
  int lane = threadIdx.x & 31;
  float* pr = PR + row * DH;
  float* pi = PI + row * DH;
  f4 a0 = *(f4*)&pr[lane * 8];
  f4 a1 = *(f4*)&pr[lane * 8 + 4];
  f4 b0 = *(f4*)&pi[lane * 8];
  f4 b1 = *(f4*)&pi[lane * 8 + 4];
  float ss = 0.f;
#pragma unroll
  for (int i = 0; i < 4; ++i)
    ss += a0[i] * a0[i] + a1[i] * a1[i] + b0[i] * b0[i] + b1[i] * b1[i];
  ss = wred32(ss);
  float inv = 1.f / fmaxf(sqrtf(ss), 1e-12f);
#pragma unroll
  for (int i = 0; i < 4; ++i) {
    a0[i] *= inv; a1[i] *= inv; b0[i] *= inv; b1[i] *= inv;
  }
  *(f4*)&pr[lane * 8] = a0;
  *(f4*)&pr[lane * 8 + 4] = a1;
  *(f4*)&pi[lane * 8] = b0;
  *(f4*)&pi[lane * 8 + 4] = b1;
}

// ------ K5: fused attention: re/im GEMM -> norm -> @values -> @ow -> LN1 ------
#define ATT_BASE (2 * 16 * ENC_LDA)  // 8320: attn/raw region
#define RED_BASE (3 * 16 * ENC_LDA)  // 12480: reduction scratch
#define T_LDA 1028                   // 1028 % 64 == 4
__global__ __launch_bounds__(256) void attn_kernel(
    const float* __restrict__ PR, const float* __restrict__ PI,
    const float* __restrict__ P, const float* __restrict__ MrT,
    const float* __restrict__ MiT, const float* __restrict__ values,
    const float* __restrict__ ow, const float* __restrict__ ob,
    const float* __restrict__ h, const float* __restrict__ n1g,
    const float* __restrict__ n1b, float* __restrict__ h1) {
  __shared__ float lds[16 * T_LDA];  // 16448 floats, regions re-used per phase
  int row0 = blockIdx.x * 16;
  int w = threadIdx.x >> 5;
  int lane = threadIdx.x & 31;
  int n = lane & 15;
  int mb = (lane >> 4) << 3;

  // stage PR/PI tiles (16 x 256 each) via async global->LDS
  for (int i = 0; i < 4; ++i) {
    int e = (threadIdx.x << 2) + i * 1024;
    int r = e >> 8, c = e & 255;
    async_ld_b128(&lds[r * ENC_LDA + c], &PR[(row0 + r) * DH + c]);
    async_ld_b128(&lds[16 * ENC_LDA + r * ENC_LDA + c],
                  &PI[(row0 + r) * DH + c]);
  }
  wait_async0();
  __syncthreads();

  // phase a: re/im GEMMs (2 s-tiles per wave). im = PR@MiT + (-PI)@MrT
  int s0 = w * 32;
  v8f are[2] = {}, aim[2] = {};
  for (int kk = 0; kk < DH; kk += 4) {
    v2f aR = fragA(lds, ENC_LDA, kk);
    v2f aI = fragA(lds + 16 * ENC_LDA, ENC_LDA, kk);
    v2f aIn;
    aIn.x = -aI.x;
    aIn.y = -aI.y;
#pragma unroll
    for (int t = 0; t < 2; ++t) {
      int col = s0 + t * 16;
      v2f bR = fragB(MrT + kk * SS + col, SS);
      v2f bI = fragB(MiT + kk * SS + col, SS);
      are[t] = wmma4(aR, bR, are[t]);
      are[t] = wmma4(aI, bI, are[t]);
      aim[t] = wmma4(aR, bI, aim[t]);
      aim[t] = wmma4(aIn, bR, aim[t]);
    }
  }
  // phase b: raw = (1-p)*|ov|^2 + p/dH, then row-normalize in LDS
#pragma unroll
  for (int t = 0; t < 2; ++t) {
    int col = s0 + t * 16 + n;
#pragma unroll
    for (int r = 0; r < 8; ++r) {
      float p = P[row0 + mb + r];
      float ov = are[t][r] * are[t][r] + aim[t][r] * aim[t][r];
      lds[ATT_BASE + (mb + r) * ENC_LDA + col] =
          (1.f - p) * ov + p * (1.f / DH);
    }
  }
  __syncthreads();
  {
    int rrow = threadIdx.x >> 4, seg = threadIdx.x & 15;
    float s = 0.f;
#pragma unroll
    for (int i = 0; i < 16; ++i)
      s += lds[ATT_BASE + rrow * ENC_LDA + seg * 16 + i];
    lds[RED_BASE + rrow * 16 + seg] = s;
  }
  __syncthreads();
  if (threadIdx.x < 16) {
    float tot = 0.f;
#pragma unroll
    for (int i = 0; i < 16; ++i) tot += lds[RED_BASE + threadIdx.x * 16 + i];
    lds[RED_BASE + 256 + threadIdx.x] = 1.f / (tot + 1e-8f);
  }
  __syncthreads();
  {
    int rrow = threadIdx.x >> 4, seg = threadIdx.x & 15;
    float inv = lds[RED_BASE + 256 + rrow];
#pragma unroll
    for (int i = 0; i < 16; ++i)
      lds[ATT_BASE + rrow * ENC_LDA + seg * 16 + i] *= inv;
  }
  __syncthreads();

  // phase c: T = attn @ values (16x1024), 8 col-tiles per wave
  int cb = w * 128;
  v8f acc[8] = {};
  for (int kk = 0; kk < SS; kk += 4) {
    v2f a = fragA(lds + ATT_BASE, ENC_LDA, kk);
#pragma unroll
    for (int t = 0; t < 8; ++t) {
      v2f b = fragB(values + kk * DD + cb + t * 16, DD);
      acc[t] = wmma4(a, b, acc[t]);
    }
  }
  __syncthreads();  // done reading attn; overwrite LDS with T
#pragma unroll
  for (int t = 0; t < 8; ++t)
#pragma unroll
    for (int r = 0; r < 8; ++r)
      lds[(mb + r) * T_LDA + cb + t * 16 + n] = acc[t][r];
  __syncthreads();

  // phase d: attn_out = T @ ow (K=1024)
  v8f acc2[8] = {};
  for (int kk = 0; kk < DD; kk += 4) {
    v2f a = fragA(lds, T_LDA, kk);
#pragma unroll
    for (int t = 0; t < 8; ++t) {
      v2f b = fragB(ow + kk * DD + cb + t * 16, DD);
      acc2[t] = wmma4(a, b, acc2[t]);
    }
  }
  // epilogue: +ob +h, LayerNorm(n1) -> h1 (d_out)
  float ps[8], pss[8];
#pragma unroll
  for (int r = 0; r < 8; ++r) { ps[r] = 0.f; pss[r] = 0.f; }
#pragma unroll
  for (int t = 0; t < 8; ++t) {
    int col = cb + t * 16 + n;
    float obv = ob[col];
#pragma unroll
    for (int r = 0; r < 8; ++r) {
      float v = acc2[t][r] + obv + h[(row0 + mb + r) * DD + col];
      acc2[t][r] = v;
      ps[r] += v;
      pss[r] += v * v;
    }
  }
#pragma unroll
  for (int r = 0; r < 8; ++r) { ps[r] = hred16(ps[r]); pss[r] = hred16(pss[r]); }
  __syncthreads();  // T dead; reuse LDS for LN reduction
  if (n == 0) {     // lanes 0 and 16 hold half-wave sums
#pragma unroll
    for (int r = 0; r < 8; ++r) {
      lds[(mb + r) * 8 + w] = ps[r];
      lds[128 + (mb + r) * 8 + w] = pss[r];
    }
  }
  __syncthreads();
  if (threadIdx.x < 16) {
    float s = 0.f, q = 0.f;
#pragma unroll
    for (int i = 0; i < 8; ++i) {
      s += lds[threadIdx.x * 8 + i];
      q += lds[128 + threadIdx.x * 8 + i];
    }
    float mu = s * (1.f / DD);
    lds[256 + threadIdx.x] = mu;
    lds[272 + threadIdx.x] = rsqrtf(q * (1.f / DD) - mu * mu + 1e-5f);
  }
  __syncthreads();
#pragma unroll
  for (int t = 0; t < 8; ++t) {
    int col = cb + t * 16 + n;
    float gv = n1g[col], bv = n1b[col];
#pragma unroll
    for (int r = 0; r < 8; ++r) {
      float mu = lds[256 + mb + r];
      float rs = lds[272 + mb + r];
      h1[(row0 + mb + r) * DD + col] = (acc2[t][r] - mu) * rs * gv + bv;
    }
  }
}

// -------- K6: fused FFN: gelu(h1@fw1+fb1)@fw2+fb2, +h1 residual, LN2 --------
#define H_LDA 1028
#define G_LDA 516                // 516 % 64 == 4
#define SG_BASE (16 * H_LDA)     // 16448
__global__ __launch_bounds__(256) void ffn_kernel(
    float* __restrict__ io,  // d_out: h1 in, final out (in-place per row tile)
    const float* __restrict__ fw1, const float* __restrict__ fb1,
    const float* __restrict__ fw2, const float* __restrict__ fb2,
    const float* __restrict__ n2g, const float* __restrict__ n2b) {
  __shared__ float lds[16 * H_LDA + 16 * G_LDA];  // ~98.8 KB of 320 KB LDS
  int row0 = blockIdx.x * 16;
  int w = threadIdx.x >> 5;
  int lane = threadIdx.x & 31;
  int n = lane & 15;
  int mb = (lane >> 4) << 3;

  // stage h1 tile 16x1024 once (async global->LDS)
  for (int i = 0; i < 16; ++i) {
    int e = (threadIdx.x << 2) + i * 1024;
    int r = e >> 10, c = e & 1023;
    async_ld_b128(&lds[r * H_LDA + c], &io[(row0 + r) * DD + c]);
  }
  __builtin_prefetch(fw1 + (threadIdx.x & 63) * 64, 0, 1);
  wait_async0();
  __syncthreads();

  int cb = w * 128;
  v8f acc2[8] = {};
  for (int cc = 0; cc < 8; ++cc) {  // 8 chunks of F=4096
    int gcb = cc * 512 + w * 64;
    v8f acc1[4] = {};
    for (int k = 0; k < DD; k += 4) {
      v2f a = fragA(lds, H_LDA, k);
#pragma unroll
      for (int t = 0; t < 4; ++t) {
        v2f b = fragB(fw1 + k * FF + gcb + t * 16, FF);
        acc1[t] = wmma4(a, b, acc1[t]);
      }
    }
    __syncthreads();  // previous chunk's GEMM2 reads of sG done
#pragma unroll
    for (int t = 0; t < 4; ++t) {
      int col = gcb + t * 16 + n;
      int lc = w * 64 + t * 16 + n;
      float bv = fb1[col];
#pragma unroll
      for (int r = 0; r < 8; ++r) {
        float x = acc1[t][r] + bv;  // exact GELU
        float g = 0.5f * x * (1.f + erff(x * 0.70710678118654752f));
        lds[SG_BASE + (mb + r) * G_LDA + lc] = g;
      }
    }
    __syncthreads();
    for (int kk = 0; kk < 512; kk += 4) {
      v2f a = fragA(lds + SG_BASE, G_LDA, kk);
#pragma unroll
      for (int t = 0; t < 8; ++t) {
        v2f b = fragB(fw2 + (cc * 512 + kk) * DD + cb + t * 16, DD);
        acc2[t] = wmma4(a, b, acc2[t]);
      }
    }
  }
  __syncthreads();
  // epilogue: + fb2 + h1 (LDS-resident), LayerNorm(n2), in-place store
  float ps[8], pss[8];
#pragma unroll
  for (int r = 0; r < 8; ++r) { ps[r] = 0.f; pss[r] = 0.f; }
#pragma unroll
  for (int t = 0; t < 8; ++t) {
    int col = cb + t * 16 + n;
    float bv = fb2[col];
#pragma unroll
    for (int r = 0; r < 8; ++r) {
      float v = acc2[t][r] + bv + lds[(mb + r) * H_LDA + col];
      acc2[t][r] = v;
      ps[r] += v;
      pss[r] += v * v;
    }
  }
#pragma unroll
  for (int r = 0; r < 8; ++r) { ps[r] = hred16(ps[r]); pss[r] = hred16(pss[r]); }
  __syncthreads();  // sG dead; reuse for LN reduction
  if (n == 0) {
#pragma unroll
    for (int r = 0; r < 8; ++r) {
      lds[SG_BASE + (mb + r) * 8 + w] = ps[r];
      lds[SG_BASE + 128 + (mb + r) * 8 + w] = pss[r];
    }
  }
  __syncthreads();
  if (threadIdx.x < 16) {
    float s = 0.f, q = 0.f;
#pragma unroll
    for (int i = 0; i < 8; ++i) {
      s += lds[SG_BASE + threadIdx.x * 8 + i];
      q += lds[SG_BASE + 128 + threadIdx.x * 8 + i];
    }
    float mu = s * (1.f / DD);
    lds[SG_BASE + 256 + threadIdx.x] = mu;
    lds[SG_BASE + 272 + threadIdx.x] = rsqrtf(q * (1.f / DD) - mu * mu + 1e-5f);
  }
  __syncthreads();
#pragma unroll
  for (int t = 0; t < 8; ++t) {
    int col = cb + t * 16 + n;
    float gv = n2g[col], bv = n2b[col];
#pragma unroll
    for (int r = 0; r < 8; ++r) {
      float mu = lds[SG_BASE + 256 + mb + r];
      float rs = lds[SG_BASE + 272 + mb + r];
      io[(row0 + mb + r) * DD + col] = (acc2[t][r] - mu) * rs * gv + bv;
    }
  }
}

// ---------------- launch ----------------
extern "C" void kernel_launch(void* const* d_in, const int* in_sizes, int n_in,
                              void* d_out, int out_size, void* d_ws,
                              size_t ws_size, hipStream_t stream) {
  (void)in_sizes; (void)n_in; (void)out_size; (void)ws_size;
  const float* h      = (const float*)d_in[0];
  const float* Wr     = (const float*)d_in[1];
  const float* br     = (const float*)d_in[2];
  const float* Wi     = (const float*)d_in[3];
  const float* bi     = (const float*)d_in[4];
  const float* uw1    = (const float*)d_in[5];
  const float* ub1    = (const float*)d_in[6];
  const float* ug     = (const float*)d_in[7];
  const float* ubeta  = (const float*)d_in[8];
  const float* uw2    = (const float*)d_in[9];
  const float* ub2    = (const float*)d_in[10];
  const float* gw1    = (const float*)d_in[11];
  const float* gb1    = (const float*)d_in[12];
  const float* gg     = (const float*)d_in[13];
  const float* gbeta  = (const float*)d_in[14];
  const float* gw2    = (const float*)d_in[15];
  const float* gb2    = (const float*)d_in[16];
  const float* m_real = (const float*)d_in[17];
  const float* m_imag = (const float*)d_in[18];
  const float* values = (const float*)d_in[19];
  const float* ow     = (const float*)d_in[20];
  const float* ob     = (const float*)d_in[21];
  const float* n1g    = (const float*)d_in[22];
  const float* n1b    = (const float*)d_in[23];
  const float* n2g    = (const float*)d_in[24];
  const float* n2b    = (const float*)d_in[25];
  const float* fw1    = (const float*)d_in[26];
  const float* fb1    = (const float*)d_in[27];
  const float* fw2    = (const float*)d_in[28];
  const float* fb2    = (const float*)d_in[29];

  float* ws    = (float*)d_ws;          // total ~34.1 MB of workspace
  float* PR    = ws;                    // 16384 x 256
  float* PI    = ws + 4194304;          // 16384 x 256
  float* P     = ws + 8388608;          // 16384
  float* GAMMA = ws + 8404992;          // 4 (padded 16)
  float* MRT   = ws + 8405008;          // 256 x 256 (transposed, normalized)
  float* MIT   = ws + 8470544;          // 256 x 256
  float* h1    = (float*)d_out;         // h1 staged in d_out, then final out

  ctx_gamma_kernel<<<BB, 1024, 0, stream>>>(h, gw1, gb1, gg, gbeta, gw2, gb2,
                                            GAMMA);
  gate_p_kernel<<<RR / 8, 256, 0, stream>>>(h, uw1, ub1, ug, ubeta, uw2, ub2,
                                            GAMMA, P);
  mnorm_kernel<<<SS / 8, 256, 0, stream>>>(m_real, m_imag, MRT, MIT);
  enc_kernel<<<RR / 16, 256, 0, stream>>>(h, Wr, br, Wi, bi, PR, PI);
  psinorm_kernel<<<RR / 8, 256, 0, stream>>>(PR, PI);
  attn_kernel<<<RR / 16, 256, 0, stream>>>(PR, PI, P, MRT, MIT, values, ow, ob,
                                           h, n1g, n1b, h1);
  ffn_kernel<<<RR / 16, 256, 0, stream>>>(h1, fw1, fb1, fw2, fb2, n2g, n2b);
}